// ROIPool_5901285064871
// MI455X (gfx1250) — compile-verified
//
#include <hip/hip_runtime.h>
#include <float.h>
#include <stdint.h>

// ---------------------------------------------------------------------------
// Problem constants (fixed by the reference setup)
// ---------------------------------------------------------------------------
#define NB   2
#define CH   256
#define FH   50
#define FW   76
#define HW   (FH * FW)          // 3800
#define PH   7
#define PW   7
#define NBIN (PH * PW)          // 49

// ---------------------------------------------------------------------------
// CDNA5 async-to-LDS helpers (gfx1250): ASYNCcnt-tracked global->LDS copy,
// and a fused LDS read + DScnt drain (keeps async WAR hazards closed).
// ---------------------------------------------------------------------------
__device__ __forceinline__ void async_g2lds_b32(uint32_t lds_byte_off, const float* g) {
    asm volatile("global_load_async_to_lds_b32 %0, %1, off"
                 :: "v"(lds_byte_off),
                    "v"((unsigned long long)(uintptr_t)g)
                 : "memory");
}
__device__ __forceinline__ float lds_read_b32_sync(uint32_t lds_byte_off) {
    float v;
    asm volatile("ds_load_b32 %0, %1\n\t"
                 "s_wait_dscnt 0x0"
                 : "=v"(v) : "v"(lds_byte_off) : "memory");
    return v;
}

// ---------------------------------------------------------------------------
// Kernel 1: NCHW -> NHWC transpose (tiled through LDS, coalesced both ways).
// Treats feat[n] as a C x HW matrix; output is HW x C.
// ---------------------------------------------------------------------------
__global__ void roipool_transpose_kernel(const float* __restrict__ in,
                                         float* __restrict__ out) {
    __shared__ float tile[32][33];
    const int n   = blockIdx.z;
    const int hw0 = blockIdx.x * 32;
    const int c0  = blockIdx.y * 32;

    #pragma unroll
    for (int j = 0; j < 4; ++j) {
        int c  = c0 + threadIdx.y + j * 8;
        int hw = hw0 + threadIdx.x;
        if (hw < HW)
            tile[threadIdx.y + j * 8][threadIdx.x] =
                in[((size_t)n * CH + c) * HW + hw];
    }
    __syncthreads();
    #pragma unroll
    for (int j = 0; j < 4; ++j) {
        int hw = hw0 + threadIdx.y + j * 8;
        int c  = c0 + threadIdx.x;
        if (hw < HW)
            out[((size_t)n * HW + hw) * CH + c] = tile[threadIdx.x][threadIdx.y + j * 8];
    }
}

// ---------------------------------------------------------------------------
// Shared ROI/bin geometry (exactly mirrors the f32 reference math).
// ---------------------------------------------------------------------------
__device__ __forceinline__ void bin_bounds(const float* __restrict__ rois, int r,
                                           float scale, int ph, int pw,
                                           int& hs, int& he, int& ws, int& we) {
    float x1 = rois[r * 4 + 0], y1 = rois[r * 4 + 1];
    float x2 = rois[r * 4 + 2], y2 = rois[r * 4 + 3];
    float sw = floorf(x1 * scale + 0.5f);
    float sh = floorf(y1 * scale + 0.5f);
    float ew = floorf(x2 * scale + 0.5f);
    float eh = floorf(y2 * scale + 0.5f);
    float roi_w = fmaxf(ew - sw + 1.0f, 1.0f);
    float roi_h = fmaxf(eh - sh + 1.0f, 1.0f);
    float bin_w = roi_w / (float)PW;
    float bin_h = roi_h / (float)PH;
    float hs_f = fminf(fmaxf(floorf((float)ph * bin_h) + sh, 0.0f), (float)FH);
    float he_f = fminf(fmaxf(ceilf((float)(ph + 1) * bin_h) + sh, 0.0f), (float)FH);
    float ws_f = fminf(fmaxf(floorf((float)pw * bin_w) + sw, 0.0f), (float)FW);
    float we_f = fminf(fmaxf(ceilf((float)(pw + 1) * bin_w) + sw, 0.0f), (float)FW);
    hs = (int)hs_f; he = (int)he_f; ws = (int)ws_f; we = (int)we_f;
}

// ---------------------------------------------------------------------------
// Kernel 2 (fast path): one workgroup per (roi, bin); lane = channel.
// Window cells streamed through a 4-deep LDS ring via async global->LDS
// copies (ASYNCcnt pipeline) -- all reads 128B-coalesced in NHWC.
// ---------------------------------------------------------------------------
#define DEPTH 4

__global__ __launch_bounds__(CH)
void roipool_nhwc_kernel(const float* __restrict__ featT,
                         const float* __restrict__ rois,
                         const int*   __restrict__ rbatch,
                         const float* __restrict__ scale_p,
                         float* __restrict__ out) {
    __shared__ float ring[DEPTH][CH];

    const int wg  = blockIdx.x;        // r * 49 + bin
    const int r   = wg / NBIN;
    const int bin = wg - r * NBIN;
    const int ph  = bin / PW;
    const int pw  = bin - ph * PW;
    const int c   = threadIdx.x;

    const float scale = scale_p[0];
    int hs, he, ws, we;
    bin_bounds(rois, r, scale, ph, pw, hs, he, ws, we);
    const int nh = he - hs, nw = we - ws;

    const size_t out_idx = ((size_t)r * CH + c) * NBIN + bin;
    if (nh <= 0 || nw <= 0) { out[out_idx] = 0.0f; return; }

    const int n = rbatch[r];
    const float* base = featT + (((size_t)n * FH + hs) * FW + ws) * CH + c;
    const int ncell = nh * nw;

    const uint32_t my_lds = (uint32_t)(uintptr_t)&ring[0][c];  // slot k at +k*1024B

    // ---- prologue: fill the pipeline ------------------------------------
    int issued = 0, jh = 0, jw = 0;
    const float* jp = base;
    const int pro = ncell < DEPTH ? ncell : DEPTH;
    for (; issued < pro; ++issued) {
        async_g2lds_b32(my_lds + (uint32_t)(issued & (DEPTH - 1)) * (CH * 4), jp);
        ++jw; jp += CH;
        if (jw == nw) { jw = 0; ++jh; jp = base + (size_t)jh * FW * CH; }
    }

    // ---- steady state ---------------------------------------------------
    float m = -FLT_MAX;
    for (int i = 0; i < ncell; ++i) {
        if (i + DEPTH <= ncell)           // DEPTH-1 cells still in flight
            asm volatile("s_wait_asynccnt 0x3" ::: "memory");
        else                              // drain tail
            asm volatile("s_wait_asynccnt 0x0" ::: "memory");

        float v = lds_read_b32_sync(my_lds + (uint32_t)(i & (DEPTH - 1)) * (CH * 4));
        m = fmaxf(m, v);

        if (issued < ncell) {             // reuse slot just consumed
            async_g2lds_b32(my_lds + (uint32_t)(i & (DEPTH - 1)) * (CH * 4), jp);
            ++issued; ++jw; jp += CH;
            if (jw == nw) { jw = 0; ++jh; jp = base + (size_t)jh * FW * CH; }
        }
    }
    out[out_idx] = m;
}

// ---------------------------------------------------------------------------
// Fallback: classic one-thread-per-output NCHW ROI max pool (used only if
// the workspace cannot hold the NHWC copy).
// ---------------------------------------------------------------------------
__global__ void roipool_nchw_kernel(const float* __restrict__ feat,
                                    const float* __restrict__ rois,
                                    const int*   __restrict__ rbatch,
                                    const float* __restrict__ scale_p,
                                    float* __restrict__ out, int total) {
    int i = blockIdx.x * blockDim.x + threadIdx.x;
    if (i >= total) return;
    int pw = i % PW;
    int ph = (i / PW) % PH;
    int c  = (i / NBIN) % CH;
    int r  = i / (NBIN * CH);

    int hs, he, ws, we;
    bin_bounds(rois, r, scale_p[0], ph, pw, hs, he, ws, we);
    if (he <= hs || we <= ws) { out[i] = 0.0f; return; }

    const int n = rbatch[r];
    const float* f = feat + ((size_t)n * CH + c) * HW;
    float m = -FLT_MAX;
    for (int h = hs; h < he; ++h)
        for (int w = ws; w < we; ++w)
            m = fmaxf(m, f[h * FW + w]);
    out[i] = m;
}

// ---------------------------------------------------------------------------
// Host launcher
// ---------------------------------------------------------------------------
extern "C" void kernel_launch(void* const* d_in, const int* in_sizes, int n_in,
                              void* d_out, int out_size, void* d_ws, size_t ws_size,
                              hipStream_t stream) {
    const float* feat    = (const float*)d_in[0];
    const float* rois    = (const float*)d_in[1];
    const int*   rbatch  = (const int*)d_in[2];
    const float* scale_p = (const float*)d_in[3];
    float*       out     = (float*)d_out;

    const int R = in_sizes[1] / 4;
    const size_t need = (size_t)NB * HW * CH * sizeof(float);  // NHWC copy

    if (ws_size >= need) {
        float* featT = (float*)d_ws;
        dim3 tb(32, 8, 1);
        dim3 tg((HW + 31) / 32, CH / 32, NB);
        roipool_transpose_kernel<<<tg, tb, 0, stream>>>(feat, featT);
        roipool_nhwc_kernel<<<dim3(R * NBIN), dim3(CH), 0, stream>>>(
            featT, rois, rbatch, scale_p, out);
    } else {
        const int total = R * CH * NBIN;
        roipool_nchw_kernel<<<dim3((total + 255) / 256), dim3(256), 0, stream>>>(
            feat, rois, rbatch, scale_p, out, total);
    }
}